// CoAttention_23545010717018
// MI455X (gfx1250) — compile-verified
//
#include <hip/hip_runtime.h>
#include <hip/hip_bf16.h>

// CoAttention on MI455X (gfx1250, wave32).
// All GEMMs use V_WMMA_F32_16X16X4_F32 (fp32 A/B/C): the problem is HBM-bound
// (~100 MB traffic vs ~8.6 GFLOP), so we keep full fp32 precision and let the
// matrix pipe do the MACs.

#define B_    16
#define PLEN  1024
#define QLEN  256
#define DDIM  256
#define MASK_VALUE (-10000000.0f)

typedef __attribute__((ext_vector_type(2))) float v2f;
typedef __attribute__((ext_vector_type(8))) float v8f;

__device__ __forceinline__ float wave_max32(float v) {
    #pragma unroll
    for (int off = 16; off > 0; off >>= 1)
        v = fmaxf(v, __shfl_xor(v, off, 32));
    return v;
}
__device__ __forceinline__ float wave_sum32(float v) {
    #pragma unroll
    for (int off = 16; off > 0; off >>= 1)
        v += __shfl_xor(v, off, 32);
    return v;
}

// ---------------------------------------------------------------------------
// K0: sp[b,p] = <passage[b,p,:], w_p>,  sq[b,q] = <question[b,q,:], w_q>
// One wave per row; 8 waves per 256-thread block.
// ---------------------------------------------------------------------------
__global__ void k0_linproj(const float* __restrict__ P, const float* __restrict__ Q,
                           const float* __restrict__ W,
                           float* __restrict__ sp, float* __restrict__ sq) {
    const int wid  = (blockIdx.x * blockDim.x + threadIdx.x) >> 5;
    const int lane = threadIdx.x & 31;
    const int NP = B_ * PLEN;
    const int NQ = B_ * QLEN;
    if (wid >= NP + NQ) return;

    const float* row;
    const float* w;
    if (wid < NP) { row = P + (size_t)wid * DDIM; w = W; }
    else          { row = Q + (size_t)(wid - NP) * DDIM; w = W + DDIM; }

    float acc = 0.0f;
    #pragma unroll
    for (int j = lane; j < DDIM; j += 32)
        acc += row[j] * w[j];
    acc = wave_sum32(acc);
    if (lane == 0) {
        if (wid < NP) sp[wid] = acc;
        else          sq[wid - NP] = acc;
    }
}

// ---------------------------------------------------------------------------
// K1: S[b,p,q] = sp + sq + sum_d (P[p,d]*w_pq[d])*Q[q,d] + bias, masked.
// Grid (PLEN/16, QLEN/16, B), one wave per 16x16 tile, K-loop 256 in steps of 4.
// ---------------------------------------------------------------------------
__global__ void k1_score(const float* __restrict__ P, const float* __restrict__ Q,
                         const int* __restrict__ pmask, const int* __restrict__ qmask,
                         const float* __restrict__ W, const float* __restrict__ bias,
                         const float* __restrict__ sp, const float* __restrict__ sq,
                         float* __restrict__ S) {
    const int b  = blockIdx.z;
    const int p0 = blockIdx.x << 4;
    const int q0 = blockIdx.y << 4;
    const int l  = threadIdx.x;
    const int n  = l & 15;          // A: row M=n ; B: col N=n
    const int kh = l >> 4;          // K-half select

    const float* w_pq = W + 2 * DDIM;
    const float* Prow = P + ((size_t)b * PLEN + p0 + n) * DDIM;
    const float* Qrow = Q + ((size_t)b * QLEN + q0 + n) * DDIM;

    v8f c = {};
    #pragma unroll 8
    for (int k = 0; k < DDIM; k += 4) {
        const int ka = k + 2 * kh;
        v2f a, bm;
        a.x  = Prow[ka]     * w_pq[ka];
        a.y  = Prow[ka + 1] * w_pq[ka + 1];
        bm.x = Qrow[ka];
        bm.y = Qrow[ka + 1];
        c = __builtin_amdgcn_wmma_f32_16x16x4_f32(false, a, false, bm,
                                                  (short)0, c, false, false);
    }

    const float sqv = sq[b * QLEN + q0 + n] + bias[0];
    const float kq  = 1.0f - (float)qmask[b * QLEN + q0 + n];
    #pragma unroll
    for (int r = 0; r < 8; ++r) {
        const int p = p0 + r + 8 * kh;           // C/D layout: row = r + 8*half
        float sc = c[r] + sp[b * PLEN + p] + sqv;
        const float kp   = 1.0f - (float)pmask[b * PLEN + p];
        const float keep = kp * kq;
        sc = sc * keep + (1.0f - keep) * MASK_VALUE;
        S[((size_t)b * PLEN + p) * QLEN + q0 + n] = sc;
    }
}

// ---------------------------------------------------------------------------
// K2a: row softmax stats over q:  rmax[b,p], rrsum[b,p] = 1/sum(exp)
// ---------------------------------------------------------------------------
__global__ void k2_rowstats(const float* __restrict__ S,
                            float* __restrict__ rmax, float* __restrict__ rrsum) {
    const int r    = (blockIdx.x * blockDim.x + threadIdx.x) >> 5;
    const int lane = threadIdx.x & 31;
    if (r >= B_ * PLEN) return;
    const float* Srow = S + (size_t)r * QLEN;

    float m = -3.4e38f;
    #pragma unroll
    for (int j = lane; j < QLEN; j += 32) m = fmaxf(m, Srow[j]);
    m = wave_max32(m);

    float s = 0.0f;
    #pragma unroll
    for (int j = lane; j < QLEN; j += 32) s += __expf(Srow[j] - m);
    s = wave_sum32(s);

    if (lane == 0) { rmax[r] = m; rrsum[r] = 1.0f / s; }
}

// ---------------------------------------------------------------------------
// K2b: column softmax stats over p:  cmax[b,q], crsum[b,q] = 1/sum(exp)
// ---------------------------------------------------------------------------
__global__ void k2_colstats(const float* __restrict__ S,
                            float* __restrict__ cmax, float* __restrict__ crsum) {
    const int cid  = (blockIdx.x * blockDim.x + threadIdx.x) >> 5;
    const int lane = threadIdx.x & 31;
    if (cid >= B_ * QLEN) return;
    const int b = cid / QLEN;
    const int q = cid % QLEN;
    const float* base = S + (size_t)b * PLEN * QLEN + q;

    float m = -3.4e38f;
    for (int j = lane; j < PLEN; j += 32) m = fmaxf(m, base[(size_t)j * QLEN]);
    m = wave_max32(m);

    float s = 0.0f;
    for (int j = lane; j < PLEN; j += 32) s += __expf(base[(size_t)j * QLEN] - m);
    s = wave_sum32(s);

    if (lane == 0) { cmax[cid] = m; crsum[cid] = 1.0f / s; }
}

// ---------------------------------------------------------------------------
// K3: q2p_att[b,q,d] = sum_p q2p[b,q,p] * passage[b,p,d]
// q2p weight built on the fly: exp(S[p,q]-cmax_q)*crsum_q*keep_q.
// (k-varying keep_p is exact to skip: exp(MASK - cmax) == 0 in fp32.)
// Grid (QLEN/16, DDIM/16, B), one wave per tile, K-loop 1024.
// ---------------------------------------------------------------------------
__global__ void k3_q2p_gemm(const float* __restrict__ S, const float* __restrict__ P,
                            const int* __restrict__ qmask,
                            const float* __restrict__ cmax, const float* __restrict__ crsum,
                            float* __restrict__ q2pA) {
    const int b  = blockIdx.z;
    const int q0 = blockIdx.x << 4;
    const int d0 = blockIdx.y << 4;
    const int l  = threadIdx.x;
    const int n  = l & 15;
    const int kh = l >> 4;

    const float cm = cmax[b * QLEN + q0 + n];
    const float kq = 1.0f - (float)qmask[b * QLEN + q0 + n];
    const float sc = crsum[b * QLEN + q0 + n] * kq;

    const float* Scol = S + (size_t)b * PLEN * QLEN + (q0 + n);   // + p*QLEN
    const float* Pb   = P + (size_t)b * PLEN * DDIM + d0 + n;     // + p*DDIM

    v8f c = {};
    #pragma unroll 4
    for (int k = 0; k < PLEN; k += 4) {
        const int ka = k + 2 * kh;
        v2f a, bm;
        a.x  = __expf(Scol[(size_t)ka * QLEN]       - cm) * sc;
        a.y  = __expf(Scol[(size_t)(ka + 1) * QLEN] - cm) * sc;
        bm.x = Pb[(size_t)ka * DDIM];
        bm.y = Pb[(size_t)(ka + 1) * DDIM];
        c = __builtin_amdgcn_wmma_f32_16x16x4_f32(false, a, false, bm,
                                                  (short)0, c, false, false);
    }

    #pragma unroll
    for (int r = 0; r < 8; ++r) {
        const int q = q0 + r + 8 * kh;
        q2pA[((size_t)b * QLEN + q) * DDIM + d0 + n] = c[r];
    }
}

// ---------------------------------------------------------------------------
// K4: fused final GEMMs sharing the p2q A-tile (built on the fly):
//   p2q_att[b,p,d] = sum_q p2q[b,p,q] * question[b,q,d]
//   coatt  [b,p,d] = sum_q p2q[b,p,q] * q2p_att[b,q,d]
// Grid (PLEN/16, DDIM/16, B), one wave per tile, two accumulators.
// ---------------------------------------------------------------------------
__global__ void k4_out_gemm(const float* __restrict__ S, const float* __restrict__ Q,
                            const float* __restrict__ q2pA,
                            const int* __restrict__ pmask,
                            const float* __restrict__ rmax, const float* __restrict__ rrsum,
                            float* __restrict__ out1, float* __restrict__ out2) {
    const int b  = blockIdx.z;
    const int p0 = blockIdx.x << 4;
    const int d0 = blockIdx.y << 4;
    const int l  = threadIdx.x;
    const int n  = l & 15;
    const int kh = l >> 4;

    const float rm = rmax[b * PLEN + p0 + n];
    const float kp = 1.0f - (float)pmask[b * PLEN + p0 + n];
    const float sc = rrsum[b * PLEN + p0 + n] * kp;

    const float* Srow = S    + ((size_t)b * PLEN + p0 + n) * QLEN;  // + k
    const float* Qb   = Q    + (size_t)b * QLEN * DDIM + d0 + n;    // + k*DDIM
    const float* Ab   = q2pA + (size_t)b * QLEN * DDIM + d0 + n;    // + k*DDIM

    v8f c1 = {};
    v8f c2 = {};
    #pragma unroll 4
    for (int k = 0; k < QLEN; k += 4) {
        const int ka = k + 2 * kh;
        v2f a, b1, b2;
        a.x  = __expf(Srow[ka]     - rm) * sc;
        a.y  = __expf(Srow[ka + 1] - rm) * sc;
        b1.x = Qb[(size_t)ka * DDIM];
        b1.y = Qb[(size_t)(ka + 1) * DDIM];
        b2.x = Ab[(size_t)ka * DDIM];
        b2.y = Ab[(size_t)(ka + 1) * DDIM];
        c1 = __builtin_amdgcn_wmma_f32_16x16x4_f32(false, a, false, b1,
                                                   (short)0, c1, false, false);
        c2 = __builtin_amdgcn_wmma_f32_16x16x4_f32(false, a, false, b2,
                                                   (short)0, c2, false, false);
    }

    #pragma unroll
    for (int r = 0; r < 8; ++r) {
        const int p = p0 + r + 8 * kh;
        const size_t o = ((size_t)b * PLEN + p) * DDIM + d0 + n;
        out1[o] = c1[r];
        out2[o] = c2[r];
    }
}

// ---------------------------------------------------------------------------
extern "C" void kernel_launch(void* const* d_in, const int* in_sizes, int n_in,
                              void* d_out, int out_size, void* d_ws, size_t ws_size,
                              hipStream_t stream) {
    const float* P    = (const float*)d_in[0];   // (B, PLEN, D)
    const float* Q    = (const float*)d_in[1];   // (B, QLEN, D)
    const int*   pm   = (const int*)  d_in[2];   // (B, PLEN)
    const int*   qm   = (const int*)  d_in[3];   // (B, QLEN)
    const float* W    = (const float*)d_in[4];   // (3D,)
    const float* bias = (const float*)d_in[5];   // (1,)
    float* out = (float*)d_out;

    // Workspace layout (floats): ~21.2 MB total
    float* S     = (float*)d_ws;                       // B*PLEN*QLEN
    float* q2pA  = S     + (size_t)B_ * PLEN * QLEN;   // B*QLEN*D
    float* sp    = q2pA  + (size_t)B_ * QLEN * DDIM;   // B*PLEN
    float* sq    = sp    + (size_t)B_ * PLEN;          // B*QLEN
    float* rmax  = sq    + (size_t)B_ * QLEN;          // B*PLEN
    float* rrsum = rmax  + (size_t)B_ * PLEN;          // B*PLEN
    float* cmax  = rrsum + (size_t)B_ * PLEN;          // B*QLEN
    float* crsum = cmax  + (size_t)B_ * QLEN;          // B*QLEN

    const int rows = B_ * PLEN + B_ * QLEN;            // 20480 rows, 8 waves/block
    k0_linproj<<<rows / 8, 256, 0, stream>>>(P, Q, W, sp, sq);

    k1_score<<<dim3(PLEN / 16, QLEN / 16, B_), 32, 0, stream>>>(
        P, Q, pm, qm, W, bias, sp, sq, S);

    k2_rowstats<<<(B_ * PLEN) / 8, 256, 0, stream>>>(S, rmax, rrsum);
    k2_colstats<<<(B_ * QLEN) / 8, 256, 0, stream>>>(S, cmax, crsum);

    k3_q2p_gemm<<<dim3(QLEN / 16, DDIM / 16, B_), 32, 0, stream>>>(
        S, P, qm, cmax, crsum, q2pA);

    k4_out_gemm<<<dim3(PLEN / 16, DDIM / 16, B_), 32, 0, stream>>>(
        S, Q, q2pA, pm, rmax, rrsum, out, out + (size_t)B_ * PLEN * DDIM);
}